// LazyMLPBlock_31344671326570
// MI455X (gfx1250) — compile-verified
//
#include <hip/hip_runtime.h>

// ---------------------------------------------------------------------------
// gpt-oss MoE block for MI455X (gfx1250), wave32 + V_WMMA_F32_16X16X4_F32.
// Memory-bound on fp32 expert weights (201 MB -> ~8.6us @ 23.3 TB/s), so we
// keep fp32 end-to-end and use the f32 WMMA to free issue slots for loads.
// ---------------------------------------------------------------------------

#if defined(__HIP_DEVICE_COMPILE__) && \
    !__has_builtin(__builtin_amdgcn_wmma_f32_16x16x4_f32)
#error "device pass: __builtin_amdgcn_wmma_f32_16x16x4_f32 not available"
#endif

typedef __attribute__((ext_vector_type(2))) float v2f;
typedef __attribute__((ext_vector_type(8))) float v8f;

namespace {

constexpr int kT = 1024;   // tokens
constexpr int kH = 1024;   // hidden
constexpr int kI = 2048;   // intermediate
constexpr int kE = 8;      // experts
constexpr float kLimit = 7.0f;
constexpr float kAlpha = 1.702f;
constexpr float kEps   = 1e-5f;

constexpr int kGroups    = kT / 16;   // max 16-token groups per expert (64)
constexpr int kNW        = 8;         // waves per block (2 per SIMD32)
constexpr int kTSStride  = kH + 4;    // pad 16B: lanes 0-15 banks {4m,4m+1}, 16-31 {4m+2,4m+3} -> conflict-free b64
constexpr int kActStride = kI + 4;    // same padding trick for phase-2 A reads

__device__ __forceinline__ v8f wmma4(v2f a, v2f b, v8f c) {
#if defined(__HIP_DEVICE_COMPILE__)
  // D = A(16x4 f32) * B(4x16 f32) + C(16x16 f32)
  return __builtin_amdgcn_wmma_f32_16x16x4_f32(false, a, false, b, (short)0, c,
                                               false, false);
#else
  (void)a; (void)b;
  return c;  // host pass: never executed, only parsed
#endif
}

__device__ __forceinline__ float fast_sigmoid(float z) {
#if defined(__HIP_DEVICE_COMPILE__)
  return __builtin_amdgcn_rcpf(1.f + __expf(-z));  // v_rcp_f32, ~1 ulp
#else
  return 0.f;
#endif
}

// Async memory->LDS copy (CDNA5): LDS dest address = low 32 bits of the
// generic shared pointer.
__device__ __forceinline__ void async_load_b128_to_lds(const float* gsrc,
                                                       float* lds_dst) {
#if defined(__HIP_DEVICE_COMPILE__)
  const unsigned dst = (unsigned)(unsigned long long)lds_dst;
  asm volatile("global_load_async_to_lds_b128 %0, %1, off"
               :
               : "v"(dst), "v"(gsrc)
               : "memory");
#else
  (void)gsrc; (void)lds_dst;
#endif
}

__device__ __forceinline__ void wait_asynccnt0() {
#if defined(__HIP_DEVICE_COMPILE__)
  asm volatile("s_wait_asynccnt 0x0" ::: "memory");
#endif
}

} // namespace

// ---------------------------------------------------------------------------
// Kernel 0: zero per-expert slot counters (ws is not re-poisoned per replay).
// ---------------------------------------------------------------------------
__global__ void moe_zero_counts(int* __restrict__ counts) {
  if (threadIdx.x < kE) counts[threadIdx.x] = 0;
}

// ---------------------------------------------------------------------------
// Kernel 1: RMSNorm + gate + top-2 + softmax; append (token,slot) to expert
// lists. One block per token, 8 waves = one wave per expert logit.
// ---------------------------------------------------------------------------
__global__ __launch_bounds__(256) void moe_norm_gate(
    const float* __restrict__ x, const float* __restrict__ nw,
    const float* __restrict__ gw, float* __restrict__ t,
    int* __restrict__ counts, int* __restrict__ tok_list,
    float* __restrict__ tok_w) {
  __shared__ float ts[kH];
  __shared__ float red[8];
  __shared__ float logits[kE];

  const int tok  = blockIdx.x;
  const int tid  = threadIdx.x;
  const int lane = tid & 31;
  const int wave = tid >> 5;

  // sum of squares (each thread owns one contiguous float4)
  const float4 xv = *(const float4*)&x[(size_t)tok * kH + tid * 4];
  float ss = xv.x * xv.x + xv.y * xv.y + xv.z * xv.z + xv.w * xv.w;
#pragma unroll
  for (int off = 16; off > 0; off >>= 1) ss += __shfl_down(ss, off, 32);
  if (lane == 0) red[wave] = ss;
  __syncthreads();
  if (tid == 0) {
    float s = 0.f;
#pragma unroll
    for (int i = 0; i < 8; ++i) s += red[i];
    red[0] = rsqrtf(s / (float)kH + kEps);
  }
  __syncthreads();
  const float inv = red[0];

  const float4 wv = *(const float4*)&nw[tid * 4];
  float4 tv;
  tv.x = xv.x * inv * wv.x;
  tv.y = xv.y * inv * wv.y;
  tv.z = xv.z * inv * wv.z;
  tv.w = xv.w * inv * wv.w;
  *(float4*)&ts[tid * 4] = tv;
  *(float4*)&t[(size_t)tok * kH + tid * 4] = tv;
  __syncthreads();

  // gate logit for expert `wave`
  float p = 0.f;
  const float* gr = gw + (size_t)wave * kH;
  for (int h = lane; h < kH; h += 32) p += ts[h] * gr[h];
#pragma unroll
  for (int off = 16; off > 0; off >>= 1) p += __shfl_down(p, off, 32);
  if (lane == 0) logits[wave] = p;
  __syncthreads();

  if (tid == 0) {
    // top-2, first-index-wins ties (matches jax.lax.top_k)
    int i0 = 0;
    float v0 = logits[0];
#pragma unroll
    for (int i = 1; i < kE; ++i)
      if (logits[i] > v0) { v0 = logits[i]; i0 = i; }
    int i1 = -1;
    float v1 = -3.4e38f;
#pragma unroll
    for (int i = 0; i < kE; ++i)
      if (i != i0 && logits[i] > v1) { v1 = logits[i]; i1 = i; }
    const float e1 = __expf(v1 - v0);      // <= 1
    const float rn = 1.f / (1.f + e1);     // precise: routing weight scales all
    const int p0 = atomicAdd(&counts[i0], 1);
    tok_list[i0 * kT + p0] = tok * 2 + 0;  // slot 0
    tok_w[i0 * kT + p0]    = rn;
    const int p1 = atomicAdd(&counts[i1], 1);
    tok_list[i1 * kT + p1] = tok * 2 + 1;  // slot 1
    tok_w[i1 * kT + p1]    = e1 * rn;
  }
}

// ---------------------------------------------------------------------------
// Kernel 2: per-(expert, 16-token group) FFN with f32 WMMA.
//   phase 1: h1 = t @ W1[e]^T + b1 ; SwiGLU -> act (fp32 in LDS, 16 x I)
//   phase 2: h2 = act @ W2[e]^T + b2 ; buf[slot] = h2 * routing_weight
// 8 waves / block (2 per SIMD32); ~206 KB dynamic LDS (CDNA5 WGP: 320 KB).
// ---------------------------------------------------------------------------
__global__ __launch_bounds__(256) void moe_ffn(
    const float* __restrict__ t, const int* __restrict__ counts,
    const int* __restrict__ tok_list, const float* __restrict__ tok_w,
    const float* __restrict__ w1, const float* __restrict__ b1,
    const float* __restrict__ w2, const float* __restrict__ b2,
    float* __restrict__ buf) {
  extern __shared__ float smem[];
  float* tS       = smem;                        // 16 * kTSStride
  float* actS     = tS + 16 * kTSStride;         // 16 * kActStride
  float* stageAll = actS + 16 * kActStride;      // kNW waves * 16 * 18
  float* rowW     = stageAll + kNW * 16 * 18;    // 16
  int*   rowTok   = (int*)(rowW + 16);           // 16 (token*2+slot, -1 = pad)

  const int e    = blockIdx.x >> 6;   // / kGroups
  const int g    = blockIdx.x & 63;
  const int cnt  = counts[e];
  const int base = g * 16;
  if (base >= cnt) return;
  const int rows = (cnt - base < 16) ? (cnt - base) : 16;

  const int tid  = threadIdx.x;
  const int lane = tid & 31;
  const int wave = tid >> 5;
  const int ln   = lane & 15;          // N (B/C/D) or M (A) index
  const int kh   = (lane >> 4) << 1;   // K sub-offset within 4: 0 or 2

  if (tid < 16) {
    rowTok[tid] = (tid < rows) ? tok_list[e * kT + base + tid] : -1;
    rowW[tid]   = (tid < rows) ? tok_w[e * kT + base + tid] : 0.f;
  }
  __syncthreads();

  // stage the 16 normalized-token rows into LDS (zeros for pad rows).
  // Real rows use the CDNA5 async memory->LDS path (no VGPR bounce).
  for (int m = wave; m < 16; m += kNW) {
    const int entry = rowTok[m];
    if (entry >= 0) {
      const float* src = t + (size_t)(entry >> 1) * kH;
      for (int h = lane * 4; h < kH; h += 128)
        async_load_b128_to_lds(src + h, &tS[m * kTSStride + h]);
    } else {
      for (int h = lane * 4; h < kH; h += 128)
        *(float4*)&tS[m * kTSStride + h] = make_float4(0.f, 0.f, 0.f, 0.f);
    }
  }
  wait_asynccnt0();
  __syncthreads();

  // ---------------- phase 1: mlp1 + interleaved SwiGLU ----------------
  float* stage = stageAll + wave * (16 * 18);
  const size_t w1e = (size_t)e * (2 * kI) * kH;
  for (int nt = wave; nt < (2 * kI) / 16; nt += kNW) {
    const int n0 = nt * 16;
    const float* brow = w1 + w1e + (size_t)(n0 + ln) * kH + kh;
    const float* arow = &tS[ln * kTSStride + kh];
    v8f c = {0.f, 0.f, 0.f, 0.f, 0.f, 0.f, 0.f, 0.f};
#pragma unroll 8
    for (int k = 0; k < kH; k += 4) {
      const v2f a = *(const v2f*)(arow + k);   // ds_load_b64 (A 16x4 layout)
      const v2f b = *(const v2f*)(brow + k);   // global_load_b64 (B 4x16 layout)
      c = wmma4(a, b, c);
    }
    const float bias = b1[(size_t)e * (2 * kI) + n0 + ln];
    // scatter D tile to per-wave staging (row m = r + 8*(lane>>4), col = ln)
#pragma unroll
    for (int r = 0; r < 8; ++r)
      stage[(r + ((lane >> 4) << 3)) * 18 + ln] = c[r] + bias;
    __builtin_amdgcn_wave_barrier();  // LDS is in-order per wave; fence compiler
    // SwiGLU: even cols = glu, odd cols = lin; 128 outputs / 32 lanes
#pragma unroll
    for (int q = 0; q < 4; ++q) {
      const int idx = lane + q * 32;
      const int m = idx & 15;
      const int j = idx >> 4;                  // 0..7
      float glu = stage[m * 18 + 2 * j];
      float lin = stage[m * 18 + 2 * j + 1];
      glu = fminf(glu, kLimit);
      lin = fminf(fmaxf(lin, -kLimit), kLimit);
      actS[m * kActStride + (n0 >> 1) + j] =
          glu * fast_sigmoid(kAlpha * glu) * (lin + 1.f);
    }
    __builtin_amdgcn_wave_barrier();
  }
  __syncthreads();

  // ---------------- phase 2: mlp2 + routing weight -> buf ----------------
  const size_t w2e = (size_t)e * kH * kI;
  for (int nt = wave; nt < kH / 16; nt += kNW) {
    const int n0 = nt * 16;
    const float* brow = w2 + w2e + (size_t)(n0 + ln) * kI + kh;
    const float* arow = &actS[ln * kActStride + kh];
    v8f c = {0.f, 0.f, 0.f, 0.f, 0.f, 0.f, 0.f, 0.f};
#pragma unroll 8
    for (int k = 0; k < kI; k += 4) {
      const v2f a = *(const v2f*)(arow + k);
      const v2f b = *(const v2f*)(brow + k);
      c = wmma4(a, b, c);
    }
    const float bias = b2[(size_t)e * kH + n0 + ln];
#pragma unroll
    for (int r = 0; r < 8; ++r) {
      const int m = r + ((lane >> 4) << 3);
      if (m < rows) {  // divergence only around stores, never around WMMA
        const int entry = rowTok[m];
        buf[(size_t)entry * kH + n0 + ln] = (c[r] + bias) * rowW[m];
      }
    }
  }
}

// ---------------------------------------------------------------------------
// Kernel 3: deterministic combine: out = x + buf[slot0] + buf[slot1].
// ---------------------------------------------------------------------------
__global__ __launch_bounds__(256) void moe_combine(
    const float* __restrict__ x, const float* __restrict__ buf,
    float* __restrict__ out) {
  const int i  = blockIdx.x * 256 + threadIdx.x;
  const int i4 = i * 4;
  const int tok = i4 >> 10;        // / kH
  const int h   = i4 & (kH - 1);
  const float4 xv = *(const float4*)&x[i4];
  const float4 a  = *(const float4*)&buf[(size_t)(2 * tok + 0) * kH + h];
  const float4 b  = *(const float4*)&buf[(size_t)(2 * tok + 1) * kH + h];
  float4 o;
  o.x = xv.x + a.x + b.x;
  o.y = xv.y + a.y + b.y;
  o.z = xv.z + a.z + b.z;
  o.w = xv.w + a.w + b.w;
  *(float4*)&out[i4] = o;
}

// ---------------------------------------------------------------------------
extern "C" void kernel_launch(void* const* d_in, const int* in_sizes, int n_in,
                              void* d_out, int out_size, void* d_ws,
                              size_t ws_size, hipStream_t stream) {
  const float* x  = (const float*)d_in[0];
  const float* nw = (const float*)d_in[1];
  const float* gw = (const float*)d_in[2];
  const float* w1 = (const float*)d_in[3];
  const float* b1 = (const float*)d_in[4];
  const float* w2 = (const float*)d_in[5];
  const float* b2 = (const float*)d_in[6];
  float* out = (float*)d_out;

  // workspace layout (bytes)
  char* ws = (char*)d_ws;
  float* t_buf   = (float*)ws;                                   // 4 MB
  float* slotbuf = (float*)(ws + (size_t)kT * kH * 4);           // 8 MB (T*2*H)
  char*  tail    = ws + (size_t)kT * kH * 4 + (size_t)kT * 2 * kH * 4;
  int*   counts   = (int*)tail;                                  // 32 B (pad 256)
  int*   tok_list = (int*)(tail + 256);                          // E*T*4
  float* tok_w    = (float*)(tail + 256 + (size_t)kE * kT * 4);  // E*T*4

  moe_zero_counts<<<1, 32, 0, stream>>>(counts);
  moe_norm_gate<<<kT, 256, 0, stream>>>(x, nw, gw, t_buf, counts, tok_list,
                                        tok_w);
  const size_t shmem =
      (size_t)(16 * kTSStride + 16 * kActStride + kNW * 16 * 18 + 16) * 4 +
      16 * 4;
  moe_ffn<<<kE * kGroups, 256, shmem, stream>>>(t_buf, counts, tok_list, tok_w,
                                                w1, b1, w2, b2, slotbuf);
  moe_combine<<<(kT * kH / 4) / 256, 256, 0, stream>>>(x, slotbuf, out);
}